// AttentionBlock_39221641347326
// MI455X (gfx1250) — compile-verified
//
#include <hip/hip_runtime.h>
#include <stdint.h>

// MI455X fused attention block (fp32 WMMA 16x16x4 flash attention).
// B=32, F=2048, hw=49, A=1.
//   k1: qkv projection -> Q [B,F,52], K [B,F,52], Vt [B,64,F]
//   k2: double-buffered flash attention + A=1 output projection
//   k3: BatchNorm1d training-stats epilogue

typedef __attribute__((ext_vector_type(2))) float v2f;
typedef __attribute__((ext_vector_type(8))) float v8f;

#define NB    32
#define NF    2048
#define NHW   49
#define DQ    52      // K-dim padded to multiple of 4
#define DV    64      // V d-dim padded to multiple of 16
#define CHUNK 64      // key/value columns per LDS chunk
#define NCHUNK (NF / CHUNK)
#define SKW   52      // sK row stride (floats): banks (52n+c)%64 conflict-free
#define SVW   68      // sV row stride: 272B rows (16B aligned), banks (4n+c)%64
#define SPW   68      // sP row stride: same padding

// ---------------------------------------------------------------------------
// CDNA5 async global->LDS copy (ASYNCcnt-tracked, in-order completion).
// LDS byte address = low 32 bits of the generic shared pointer.
__device__ __forceinline__ void async_b128(uint32_t lds_off, const float* gaddr) {
    asm volatile("global_load_async_to_lds_b128 %0, %1, off"
                 :: "v"(lds_off), "v"((unsigned long long)(uintptr_t)gaddr)
                 : "memory");
}

// ---------------------------------------------------------------------------
// Kernel 1: qkv = xf @ W_qkv^T ; scatter into Q/K (row-major, K-padded) and
// Vt (d-major so k2 can async-copy contiguous [d][g] rows). Pads written as 0.
__global__ __launch_bounds__(256) void qkv_kernel(
    const float* __restrict__ x, const float* __restrict__ Wqkv,
    float* __restrict__ Q, float* __restrict__ K, float* __restrict__ Vt)
{
    __shared__ float sW[147 * 49];
    __shared__ float sX[8][49];
    const int tid = threadIdx.x;
    const size_t row0 = (size_t)blockIdx.x * 8;

    for (int idx = tid; idx < 147 * 49; idx += 256) sW[idx] = Wqkv[idx];
    for (int idx = tid; idx < 8 * 49; idx += 256) {
        int r = idx / 49, d = idx - r * 49;
        sX[r][d] = x[(row0 + r) * 49 + d];
    }
    __syncthreads();

    // idx = e*8 + r so groups of 8 lanes share e -> coalesced Vt writes
    for (int idx = tid; idx < 147 * 8; idx += 256) {
        int r = idx & 7, e = idx >> 3;
        size_t i = row0 + r;
        int b = (int)(i >> 11), f = (int)(i & 2047);
        float acc = 0.f;
        #pragma unroll
        for (int d = 0; d < 49; ++d) acc += sX[r][d] * sW[e * 49 + d];
        if (e < 49)       Q[i * DQ + e] = acc;
        else if (e < 98)  K[i * DQ + (e - 49)] = acc;
        else              Vt[((size_t)b * DV + (e - 98)) * NF + f] = acc;
    }
    for (int idx = tid; idx < 21 * 8; idx += 256) {
        int r = idx & 7, p = idx >> 3;
        size_t i = row0 + r;
        int b = (int)(i >> 11), f = (int)(i & 2047);
        if (p < 3)       Q[i * DQ + 49 + p] = 0.f;
        else if (p < 6)  K[i * DQ + 49 + (p - 3)] = 0.f;
        else             Vt[((size_t)b * DV + 49 + (p - 6)) * NF + f] = 0.f;
    }
}

// ---------------------------------------------------------------------------
// Kernel 2: fused flash attention, double-buffered async K/V chunks.
// 256 threads = 8 waves; wave w owns query rows [blk*128 + w*16, +16).
__global__ __launch_bounds__(256) void attn_kernel(
    const float* __restrict__ Q, const float* __restrict__ K,
    const float* __restrict__ Vt, const float* __restrict__ Wout,
    const float* __restrict__ bout, float* __restrict__ outpre)
{
    __shared__ __align__(16) float sK[2][CHUNK][SKW];  // [g][d]  (B-frag pairs contiguous)
    __shared__ __align__(16) float sV[2][DV][SVW];     // [d][g]  (B-frag pairs contiguous)
    __shared__ __align__(16) float sP[8][16][SPW];     // per-wave probs tile

    const int tid  = threadIdx.x;
    const int wave = tid >> 5;
    const int lane = tid & 31;
    const int hl   = lane >> 4;
    const int ln   = lane & 15;
    const int b    = blockIdx.y;
    const int rowbase = blockIdx.x * 128 + wave * 16;

    const float* kbase  = K  + (size_t)b * NF * DQ;
    const float* vtbase = Vt + (size_t)b * DV * NF;

    // --- Q A-fragments (ISA 32-bit A 16x4 layout), softmax scale folded in ---
    const float scale = 1.0f / 7.0f;
    const float* qrow = Q + ((size_t)(b * NF + rowbase + ln)) * DQ;
    v2f aq[13];
    #pragma unroll
    for (int k = 0; k < 13; ++k) {
        int d = 4 * k + 2 * hl;
        aq[k].x = qrow[d]     * scale;
        aq[k].y = qrow[d + 1] * scale;
    }

    float m_run[8], s_run[8];
    v8f acc[4];
    #pragma unroll
    for (int j = 0; j < 8; ++j) { m_run[j] = -1e30f; s_run[j] = 0.f; }
    #pragma unroll
    for (int t = 0; t < 4; ++t) acc[t] = {};

    // per-wave async issue counts are uniform (waves 0-1: 8, waves 2-7: 7)
    auto load_chunk = [&](int buf, int g0) {
        for (int idx = tid; idx < CHUNK * (DQ / 4); idx += 256) {   // 832 x b128
            int g = idx / 13, q = idx - g * 13;
            async_b128((uint32_t)(uintptr_t)&sK[buf][g][q * 4],
                       kbase + (size_t)(g0 + g) * DQ + q * 4);
        }
        for (int idx = tid; idx < DV * (CHUNK / 4); idx += 256) {   // 1024 x b128
            int d = idx >> 4, q = idx & 15;
            async_b128((uint32_t)(uintptr_t)&sV[buf][d][q * 4],
                       vtbase + (size_t)d * NF + g0 + q * 4);
        }
    };

    load_chunk(0, 0);   // prologue

    for (int ch = 0; ch < NCHUNK; ++ch) {
        const int cur = ch & 1;
        __syncthreads();                    // buffer cur^1 fully consumed (ch-1)
        if (ch + 1 < NCHUNK) {
            load_chunk(cur ^ 1, (ch + 1) * CHUNK);
            // in-order completion: <=7 outstanding => chunk `cur` fully landed
            asm volatile("s_wait_asynccnt 0x7" ::: "memory");
        } else {
            asm volatile("s_wait_asynccnt 0x0" ::: "memory");
        }
        __syncthreads();                    // cur chunk visible to all waves

        // --- S = (Q*scale) @ K^T : k-outer / tile-inner so the 4 independent
        //     accumulator chains interleave (WMMA dep distance 4, not 1) ---
        v8f sc[4];
        #pragma unroll
        for (int t = 0; t < 4; ++t) sc[t] = {};
        #pragma unroll
        for (int k = 0; k < 13; ++k) {
            #pragma unroll
            for (int t = 0; t < 4; ++t) {
                v2f bf = *(const v2f*)&sK[cur][t * 16 + ln][4 * k + 2 * hl];
                sc[t] = __builtin_amdgcn_wmma_f32_16x16x4_f32(
                            false, aq[k], false, bf, (short)0, sc[t], false, false);
            }
        }

        // --- online softmax (row stats replicated in each 16-lane half) ---
        #pragma unroll
        for (int j = 0; j < 8; ++j) {
            float mx = fmaxf(fmaxf(sc[0][j], sc[1][j]), fmaxf(sc[2][j], sc[3][j]));
            #pragma unroll
            for (int off = 8; off; off >>= 1) mx = fmaxf(mx, __shfl_xor(mx, off));
            float mnew  = fmaxf(m_run[j], mx);
            float alpha = __expf(m_run[j] - mnew);
            m_run[j] = mnew;
            float ls = 0.f;
            #pragma unroll
            for (int t = 0; t < 4; ++t) {
                float p = __expf(sc[t][j] - mnew);
                sc[t][j] = p;
                ls += p;
            }
            #pragma unroll
            for (int off = 8; off; off >>= 1) ls += __shfl_xor(ls, off);
            s_run[j] = s_run[j] * alpha + ls;
            #pragma unroll
            for (int t = 0; t < 4; ++t) acc[t][j] *= alpha;
        }

        // --- C-layout -> A-layout via per-wave LDS staging (conflict-free) ---
        #pragma unroll
        for (int t = 0; t < 4; ++t) {
            #pragma unroll
            for (int j = 0; j < 8; ++j)
                sP[wave][j + 8 * hl][t * 16 + ln] = sc[t][j];
        }
        __syncthreads();

        // --- acc += P @ V : kc-outer / td-inner: A-frag loaded once per kc
        //     (16 sP loads instead of 64) and accumulator chains interleave ---
        #pragma unroll
        for (int kc = 0; kc < 16; ++kc) {
            const int kb = kc * 4 + 2 * hl;
            v2f af = *(const v2f*)&sP[wave][ln][kb];
            #pragma unroll
            for (int td = 0; td < 4; ++td) {
                v2f bf = *(const v2f*)&sV[cur][td * 16 + ln][kb];
                acc[td] = __builtin_amdgcn_wmma_f32_16x16x4_f32(
                              false, af, false, bf, (short)0, acc[td], false, false);
            }
        }
    }

    // --- fused output projection: out[row] = (acc/rowsum) . W_out + b_out ---
    float w4[4];
    #pragma unroll
    for (int t = 0; t < 4; ++t) {
        int d = t * 16 + ln;
        w4[t] = (d < NHW) ? Wout[d] : 0.f;
    }
    const float b0 = bout[0];
    #pragma unroll
    for (int j = 0; j < 8; ++j) {
        float part = acc[0][j] * w4[0] + acc[1][j] * w4[1]
                   + acc[2][j] * w4[2] + acc[3][j] * w4[3];
        #pragma unroll
        for (int off = 8; off; off >>= 1) part += __shfl_xor(part, off);
        if (ln == 0) {
            int row = rowbase + j + 8 * hl;
            outpre[b * NF + row] = part / s_run[j] + b0;
        }
    }
}

// ---------------------------------------------------------------------------
// Kernel 3: BatchNorm1d (training stats): per-channel f over 32 batch samples.
__global__ __launch_bounds__(256) void bn_kernel(
    const float* __restrict__ pre, const float* __restrict__ gamma,
    const float* __restrict__ beta, float* __restrict__ out)
{
    int f = blockIdx.x * blockDim.x + threadIdx.x;
    if (f >= NF) return;
    float vals[NB];
    float s = 0.f, s2 = 0.f;
    #pragma unroll
    for (int b = 0; b < NB; ++b) {
        float v = pre[b * NF + f];
        vals[b] = v; s += v; s2 += v * v;
    }
    float mean = s * (1.0f / NB);
    float var  = s2 * (1.0f / NB) - mean * mean;   // biased variance
    float g = gamma[f] * rsqrtf(var + 1e-5f);
    float bb = beta[f];
    #pragma unroll
    for (int b = 0; b < NB; ++b)
        out[b * NF + f] = (vals[b] - mean) * g + bb;
}

// ---------------------------------------------------------------------------
extern "C" void kernel_launch(void* const* d_in, const int* in_sizes, int n_in,
                              void* d_out, int out_size, void* d_ws, size_t ws_size,
                              hipStream_t stream) {
    const float* x     = (const float*)d_in[0];
    const float* Wqkv  = (const float*)d_in[1];
    const float* Wout  = (const float*)d_in[2];
    const float* bout  = (const float*)d_in[3];
    const float* gamma = (const float*)d_in[4];
    const float* beta  = (const float*)d_in[5];

    float* ws = (float*)d_ws;
    const size_t BF = (size_t)NB * NF;
    float* Q   = ws;                 // BF * 52
    float* K   = Q  + BF * DQ;       // BF * 52
    float* Vt  = K  + BF * DQ;       // NB * 64 * NF (== BF * 64)
    float* pre = Vt + BF * DV;       // BF

    qkv_kernel<<<(int)(BF / 8), 256, 0, stream>>>(x, Wqkv, Q, K, Vt);
    attn_kernel<<<dim3(NF / 128, NB), 256, 0, stream>>>(Q, K, Vt, Wout, bout, pre);
    bn_kernel<<<NF / 256, 256, 0, stream>>>(pre, gamma, beta, (float*)d_out);
}